// EcTransOctahedralSHTModule_3770981286263
// MI455X (gfx1250) — compile-verified
//
#include <hip/hip_runtime.h>

typedef float v2f __attribute__((ext_vector_type(2)));
typedef float v8f __attribute__((ext_vector_type(8)));
typedef unsigned int v4u __attribute__((ext_vector_type(4)));
typedef int v4i __attribute__((ext_vector_type(4)));
typedef int v8i __attribute__((ext_vector_type(8)));

#define NLAT 128      // N_LAT_NH
#define NM   128      // T+1
#define NV   128      // N_VARS
#define NB   4        // BATCH
#define NGRID 70144   // CUMS[-1]
#define CHUNK 64
#define TWO_PI 6.28318530717958647692f

__device__ __forceinline__ int lons(int i)       { return 20 + 4 * i; }
__device__ __forceinline__ int base_north(int i) { return 2 * i * i + 18 * i; }
__device__ __forceinline__ int base_south(int i) { int u = i + 1; return NGRID - (2 * u * u + 18 * u); }

// ---------------------------------------------------------------------------
// Stage 1: folded ragged DFT as WMMA GEMM.
// Per (lat, b): D = A[{cos,sin} x nlon] * B[nlon x 256 ({+},{-}) vars]
// Block = (v-group of 64 cols) x b x lat; 8 waves = 8 m-tiles.
// Each wave: 8 accumulators (4 v-tiles x {cos,sin}); one sincos A-gen
// feeds 8 v_wmma_f32_16x16x4_f32 per K-step. Integer residues keep the
// phase (m*k mod nlon) exact. Writes W[c][b][m][lat][v],
// c in {sym_re, sym_im, anti_re, anti_im}.
// ---------------------------------------------------------------------------
__global__ __launch_bounds__(256)
void sht_stage1_dft(const float* __restrict__ x, float* __restrict__ W)
{
    const int g   = blockIdx.x;        // 0,1: plus(v 0-63 / 64-127); 2,3: minus
    const int b   = blockIdx.y;
    const int lat = blockIdx.z;

    const int   nlon = lons(lat);
    const int   mcnt = min(5 + lat, 128);
    const float inv  = 1.0f / (float)nlon;

    const int tid  = threadIdx.x;
    const int wave = tid >> 5;
    const int lane = tid & 31;
    const int lh   = lane & 15;
    const int koff = (lane < 16) ? 0 : 2;

    const int   vbase = (g & 1) * 64;
    const float sgn   = (g < 2) ? 1.0f : -1.0f;

    const float* xn = x + ((size_t)b * NGRID + base_north(lat)) * NV;
    const float* xs = x + ((size_t)b * NGRID + base_south(lat)) * NV;

    __shared__ float bsT[64][CHUNK + 2];   // [v-col][k], padded (16.9 KB)

    // exact integer phase residues: r = (m * k) mod nlon, stepped by 4m
    const int   m    = wave * 16 + lh;
    const float amp  = (m < mcnt) ? 1.0f : 0.0f;     // reference zero-pads m>=mcnt
    const int   step = (4 * m) % nlon;
    int r0 = (m * koff) % nlon;
    int r1 = (m * (koff + 1)) % nlon;

    v8f accC[4], accS[4];
    #pragma unroll
    for (int t = 0; t < 4; ++t) {
        accC[t] = (v8f){0.f,0.f,0.f,0.f,0.f,0.f,0.f,0.f};
        accS[t] = (v8f){0.f,0.f,0.f,0.f,0.f,0.f,0.f,0.f};
    }

    const int nchunks = (nlon + CHUNK - 1) / CHUNK;
    for (int c = 0; c < nchunks; ++c) {
        const int kc0 = c * CHUNK;

        // prefetch next chunk (global_prefetch_b8)
        {
            int pk = kc0 + CHUNK + (tid >> 6);
            int pc = tid & 63;
            if (pk < nlon) {
                __builtin_prefetch(&xn[(size_t)pk * NV + vbase + pc], 0, 0);
                __builtin_prefetch(&xs[(size_t)pk * NV + vbase + pc], 0, 0);
            }
        }

        // cooperative hemispheric fold: 64 k x 64 v, zero-padded tail
        #pragma unroll
        for (int j = 0; j < (CHUNK * 64) / 256; ++j) {
            int idx = tid + j * 256;
            int kk  = idx >> 6;
            int cc  = idx & 63;
            int kg  = kc0 + kk;
            float val = 0.0f;
            if (kg < nlon) {
                float a = xn[(size_t)kg * NV + vbase + cc];
                float s = xs[(size_t)kg * NV + vbase + cc];
                val = a + sgn * s;
            }
            bsT[cc][kk] = val;
        }
        __syncthreads();

        #pragma unroll 2
        for (int s = 0; s < CHUNK / 4; ++s) {
            const int ks = s * 4 + koff;
            float s0, c0, s1, c1;
            __sincosf(TWO_PI * ((float)r0 * inv), &s0, &c0);
            __sincosf(TWO_PI * ((float)r1 * inv), &s1, &c1);
            v2f afc, afs;
            afc.x = c0 * amp; afc.y = c1 * amp;
            afs.x = s0 * amp; afs.y = s1 * amp;
            #pragma unroll
            for (int t = 0; t < 4; ++t) {
                v2f bf;
                bf.x = bsT[t * 16 + lh][ks];
                bf.y = bsT[t * 16 + lh][ks + 1];
                accC[t] = __builtin_amdgcn_wmma_f32_16x16x4_f32(
                    false, afc, false, bf, (short)0, accC[t], false, false);
                accS[t] = __builtin_amdgcn_wmma_f32_16x16x4_f32(
                    false, afs, false, bf, (short)0, accS[t], false, false);
            }
            r0 += step; if (r0 >= nlon) r0 -= nlon;
            r1 += step; if (r1 >= nlon) r1 -= nlon;
        }
        __syncthreads();
    }

    // store: 1/nlon scale; sin (imag) rows carry the rfft minus sign
    const int cbase = (g < 2) ? 0 : 2;
    float* WpC = W + ((size_t)(cbase + 0) * NB + b) * ((size_t)NM * NLAT * NV);
    float* WpS = W + ((size_t)(cbase + 1) * NB + b) * ((size_t)NM * NLAT * NV);
    #pragma unroll
    for (int t = 0; t < 4; ++t) {
        #pragma unroll
        for (int r = 0; r < 8; ++r) {
            int mrow = wave * 16 + r + ((lane < 16) ? 0 : 8);
            int vcol = vbase + t * 16 + lh;
            size_t idx = ((size_t)mrow * NLAT + lat) * NV + vcol;
            WpC[idx] = accC[t][r] * inv;
            WpS[idx] = accS[t][r] * (-inv);
        }
    }
}

// ---------------------------------------------------------------------------
// Stage 2: per-m Legendre WMMA GEMM.
// Block = (atype: sym/anti) x b x m; the full 64x128 A slab is DMA'd into
// LDS by the Tensor Data Mover (TDM) with pad_enable producing the
// bank-conflict-free stride-132 layout (fallback: cooperative loads).
// 8 waves = 8 v-tiles; each wave: 8 accumulators (4 l-tiles x re/im), so
// each W element is read exactly once overall. re/im pack into one float2
// store -> coalesced output. Rows interleave sym(even)/anti(odd).
// ---------------------------------------------------------------------------
__global__ __launch_bounds__(256)
void sht_stage2_legendre(const float* __restrict__ syms,
                         const float* __restrict__ anti,
                         const float* __restrict__ W,
                         float* __restrict__ out)
{
    const int atype = blockIdx.x;        // 0: symmetric, 1: antisymmetric
    const int b     = blockIdx.y;
    const int m     = blockIdx.z;

    const int tid   = threadIdx.x;
    const int wave  = tid >> 5;
    const int lane  = tid & 31;
    const int lh    = lane & 15;
    const int koff  = (lane < 16) ? 0 : 2;
    const int vbase = wave * 16;

    __shared__ float As[64][132];        // 33.8 KB; 132 = 128 + 4 pad (16B-aligned rows)

    const float* Amat = (atype ? anti : syms) + (size_t)m * (64 * 128);

#if __has_builtin(__builtin_amdgcn_tensor_load_to_lds)
    // TDM: one descriptor, whole 64x128 f32 tile, 4-DWORD pad every 128 DWORDs.
    if (tid < 32) {
        unsigned long long ga = (unsigned long long)(const void*)Amat;
        unsigned ldsa = (unsigned)(uintptr_t)(void*)&As[0][0]; // generic->LDS offset (low 32b)
        v4u g0;
        g0[0] = 1u;                                            // count=1, user descriptor
        g0[1] = ldsa;                                          // lds_addr (bytes)
        g0[2] = (unsigned)(ga & 0xffffffffu);                  // global_addr[31:0]
        g0[3] = (unsigned)((ga >> 32) & 0x01ffffffu)
              | 0x80000000u;                                   // global_addr[56:32] | type=2
        v8i g1;
        g1[0] = 0x07920000;   // data_size=4B(17:16=2), pad_en(20), interval=6(24:22), amount=3(31:25)
        g1[1] = (int)(128u << 16);   // tensor_dim0 = 128 (low 16 bits at 63:48)
        g1[2] = (int)(64u << 16);    // tensor_dim1 = 64  (low 16 bits at 95:80)
        g1[3] = (int)(128u << 16);   // tile_dim0 = 128 (bits 127:112)
        g1[4] = 64;                  // tile_dim1 = 64  (bits 143:128)
        g1[5] = 128;                 // tensor_dim0_stride = 128 (bits 191:160)
        g1[6] = 0;
        g1[7] = 0;
        v4i gz4 = {0, 0, 0, 0};            // 2D tensor: groups 2/3 unused
        v8i gz8 = {0, 0, 0, 0, 0, 0, 0, 0};
        __builtin_amdgcn_tensor_load_to_lds(g0, g1, gz4, gz4, gz8, 0);
        __builtin_amdgcn_s_wait_tensorcnt(0);
    }
    __syncthreads();
#else
    #pragma unroll
    for (int j = 0; j < 32; ++j) {
        int idx = tid + j * 256;
        As[idx >> 7][idx & 127] = Amat[idx];
    }
    __syncthreads();
#endif

    const float* B0 = W + (((size_t)(atype * 2 + 0) * NB + b) * NM + m) * ((size_t)NLAT * NV);
    const float* B1 = W + (((size_t)(atype * 2 + 1) * NB + b) * NM + m) * ((size_t)NLAT * NV);

    v8f accR[4], accI[4];
    #pragma unroll
    for (int t = 0; t < 4; ++t) {
        accR[t] = (v8f){0.f,0.f,0.f,0.f,0.f,0.f,0.f,0.f};
        accI[t] = (v8f){0.f,0.f,0.f,0.f,0.f,0.f,0.f,0.f};
    }

    #pragma unroll 2
    for (int k0 = 0; k0 < NLAT; k0 += 4) {
        const int k = k0 + koff;
        v2f bfR, bfI;
        bfR.x = B0[(size_t)k * NV + vbase + lh];
        bfR.y = B0[(size_t)(k + 1) * NV + vbase + lh];
        bfI.x = B1[(size_t)k * NV + vbase + lh];
        bfI.y = B1[(size_t)(k + 1) * NV + vbase + lh];
        #pragma unroll
        for (int lt = 0; lt < 4; ++lt) {
            v2f af;
            af.x = As[lt * 16 + lh][k];
            af.y = As[lt * 16 + lh][k + 1];
            accR[lt] = __builtin_amdgcn_wmma_f32_16x16x4_f32(
                false, af, false, bfR, (short)0, accR[lt], false, false);
            accI[lt] = __builtin_amdgcn_wmma_f32_16x16x4_f32(
                false, af, false, bfI, (short)0, accI[lt], false, false);
        }
    }

    float2* out2 = (float2*)out;
    #pragma unroll
    for (int lt = 0; lt < 4; ++lt) {
        #pragma unroll
        for (int r = 0; r < 8; ++r) {
            int ll  = lt * 16 + r + ((lane < 16) ? 0 : 8);
            int row = 2 * ll + atype;               // even = sym, odd = anti
            size_t o = (((size_t)b * 128 + row) * NM + m) * NV + (vbase + lh);
            float2 v; v.x = accR[lt][r]; v.y = accI[lt][r];
            out2[o] = v;
        }
    }
}

// ---------------------------------------------------------------------------
extern "C" void kernel_launch(void* const* d_in, const int* in_sizes, int n_in,
                              void* d_out, int out_size, void* d_ws, size_t ws_size,
                              hipStream_t stream)
{
    const float* x    = (const float*)d_in[0];   // [4,1,70144,128]
    const float* syms = (const float*)d_in[1];   // [128,64,128]
    const float* anti = (const float*)d_in[2];   // [128,64,128]
    float* out = (float*)d_out;                  // [4,1,128,128,128,2]
    float* W   = (float*)d_ws;                   // 4*4*128*128*128 fp32 = 128 MiB

    (void)in_sizes; (void)n_in; (void)out_size; (void)ws_size;

    dim3 g1(4, NB, NLAT);
    sht_stage1_dft<<<g1, 256, 0, stream>>>(x, W);

    dim3 g2(2, NB, NM);
    sht_stage2_legendre<<<g2, 256, 0, stream>>>(syms, anti, W, out);
}